// GraphAutoencoder_62234076119723
// MI455X (gfx1250) — compile-verified
//
#include <hip/hip_runtime.h>

typedef __attribute__((ext_vector_type(2))) float v2f;
typedef __attribute__((ext_vector_type(8))) float v8f;

#define FDIM 128

// ---------------- degree / normalization ----------------
__global__ void k_fill_deg(float* __restrict__ deg, int N) {
    int i = blockIdx.x * blockDim.x + threadIdx.x;
    if (i < N) deg[i] = 1.0f;  // self-loop contribution
}

__global__ void k_count_deg(const long long* __restrict__ dst,
                            float* __restrict__ deg, int E) {
    int i = blockIdx.x * blockDim.x + threadIdx.x;
    if (i < E) unsafeAtomicAdd(&deg[(int)dst[i]], 1.0f);
}

__global__ void k_rsqrt_inplace(float* __restrict__ deg, int N) {
    int i = blockIdx.x * blockDim.x + threadIdx.x;
    if (i < N) deg[i] = rsqrtf(deg[i]);  // deg >= 1 always
}

// ---------------- fp32 WMMA GEMM: C[N x 128] = A[N x 128] * W[128 x 128] ----------------
// Block = 256 threads = 8 waves. Block covers 16 rows; wave w covers cols [16w, 16w+16).
// Per wave: 32 x v_wmma_f32_16x16x4_f32 accumulating over K=128.
__global__ __launch_bounds__(256) void k_gemm_wmma(
    const float* __restrict__ A, const float* __restrict__ W,
    const float* __restrict__ bias, float* __restrict__ C,
    int N, int fuse_bias_relu)
{
    __shared__ float As[16][FDIM + 4];  // +4 pad: stride 132 -> conflict-free LDS reads

    const int row0 = blockIdx.x * 16;
    const int tid  = threadIdx.x;

    // Stage the 16x128 A strip into LDS (512 float4 loads across 256 threads).
    for (int i = tid; i < 16 * (FDIM / 4); i += 256) {
        int r  = i >> 5;       // 32 float4 per row
        int c4 = i & 31;
        float4 v = make_float4(0.f, 0.f, 0.f, 0.f);
        if (row0 + r < N)
            v = ((const float4*)(A + (size_t)(row0 + r) * FDIM))[c4];
        *(float4*)&As[r][c4 * 4] = v;
    }
    __syncthreads();

    const int lane = tid & 31;
    const int wave = tid >> 5;
    const int half = lane >> 4;   // 0: lanes 0-15, 1: lanes 16-31
    const int m    = lane & 15;
    const int col0 = wave * 16;

    v8f acc = {};
#pragma unroll
    for (int k0 = 0; k0 < FDIM; k0 += 4) {
        const int ka = k0 + 2 * half;   // ISA 16x4 f32 A layout: lanes16-31 hold K+2,K+3
        v2f a, b;
        a.x = As[m][ka];
        a.y = As[m][ka + 1];
        b.x = W[(size_t)ka * FDIM + col0 + m];
        b.y = W[(size_t)(ka + 1) * FDIM + col0 + m];
        acc = __builtin_amdgcn_wmma_f32_16x16x4_f32(
            /*neg_a=*/false, a, /*neg_b=*/false, b,
            /*c_mod=*/(short)0, acc, /*reuse_a=*/false, /*reuse_b=*/false);
    }

    // C/D layout: VGPR r -> row (r + 8*half), col = lane&15 within tile.
    const float bv = fuse_bias_relu ? bias[col0 + m] : 0.0f;
#pragma unroll
    for (int r = 0; r < 8; ++r) {
        int row = row0 + r + 8 * half;
        if (row < N) {
            float v = acc[r];
            if (fuse_bias_relu) v = fmaxf(v + bv, 0.0f);
            C[(size_t)row * FDIM + col0 + m] = v;
        }
    }
}

// ---------------- self-loop init: out[n] = h[n] * dinv[n]^2 ----------------
__global__ void k_selfloop(const float* __restrict__ h, const float* __restrict__ dinv,
                           float* __restrict__ out, int N) {
    long long i = (long long)blockIdx.x * blockDim.x + threadIdx.x;  // per float4
    long long total = (long long)N * (FDIM / 4);
    if (i >= total) return;
    int n = (int)(i >> 5);
    float di = dinv[n];
    float s = di * di;
    float4 v = ((const float4*)h)[i];
    v.x *= s; v.y *= s; v.z *= s; v.w *= s;
    ((float4*)out)[i] = v;
}

// ---------------- edge scatter: out[dst] += h[src] * dinv[src]*dinv[dst] ----------------
// One wave per edge; lane l handles float4 l (32 x 4 = 128 features).
__global__ __launch_bounds__(256) void k_scatter(
    const float* __restrict__ h, const long long* __restrict__ src,
    const long long* __restrict__ dst, const float* __restrict__ dinv,
    float* __restrict__ out, int E)
{
    int gw   = (int)(((long long)blockIdx.x * blockDim.x + threadIdx.x) >> 5);
    int lane = threadIdx.x & 31;
    if (gw >= E) return;
    int s = (int)src[gw];
    int d = (int)dst[gw];
    float nm = dinv[s] * dinv[d];
    float4 v = ((const float4*)(h + (size_t)s * FDIM))[lane];
    float* op = out + (size_t)d * FDIM + (size_t)lane * 4;
    unsafeAtomicAdd(op + 0, v.x * nm);
    unsafeAtomicAdd(op + 1, v.y * nm);
    unsafeAtomicAdd(op + 2, v.z * nm);
    unsafeAtomicAdd(op + 3, v.w * nm);
}

// ---------------- bias + relu ----------------
__global__ void k_bias_relu(const float* __restrict__ agg, const float* __restrict__ b,
                            float* __restrict__ out, int N) {
    long long i = (long long)blockIdx.x * blockDim.x + threadIdx.x;
    long long total = (long long)N * FDIM;
    if (i >= total) return;
    float v = agg[i] + b[(int)(i & (FDIM - 1))];
    out[i] = fmaxf(v, 0.0f);
}

extern "C" void kernel_launch(void* const* d_in, const int* in_sizes, int n_in,
                              void* d_out, int out_size, void* d_ws, size_t ws_size,
                              hipStream_t stream) {
    const float*     x    = (const float*)d_in[0];
    const long long* ei   = (const long long*)d_in[1];  // int64 edge_index [2, E]
    const float*     W1   = (const float*)d_in[2];
    const float*     b1   = (const float*)d_in[3];
    const float*     W2   = (const float*)d_in[4];
    const float*     b2   = (const float*)d_in[5];
    const float*     Wd   = (const float*)d_in[6];
    const float*     bd   = (const float*)d_in[7];

    const int N = in_sizes[0] / FDIM;
    const int E = in_sizes[1] / 2;
    const long long* srcp = ei;
    const long long* dstp = ei + E;

    char* ws = (char*)d_ws;
    size_t o = 0;
    float* dinv = (float*)(ws + o); o += (((size_t)N * 4) + 255) & ~(size_t)255;
    float* bufH = (float*)(ws + o); o += (size_t)N * FDIM * 4;
    float* bufA = (float*)(ws + o); o += (size_t)N * FDIM * 4;
    float* out  = (float*)d_out;

    dim3 b256(256);
    int gN    = (N + 255) / 256;
    int gE    = (E + 255) / 256;
    int gEw   = (int)((((long long)E * 32) + 255) / 256);  // wave per edge
    int gNF4  = (int)((((long long)N * (FDIM / 4)) + 255) / 256);
    int gNF   = (int)((((long long)N * FDIM) + 255) / 256);
    int gGemm = (N + 15) / 16;

    // normalization: dinv = rsqrt(1 + in-degree)
    k_fill_deg<<<gN, b256, 0, stream>>>(dinv, N);
    k_count_deg<<<gE, b256, 0, stream>>>(dstp, dinv, E);
    k_rsqrt_inplace<<<gN, b256, 0, stream>>>(dinv, N);

    // ---- layer 1: h1 = relu(Agg(x @ W1) + b1) ----
    k_gemm_wmma<<<gGemm, b256, 0, stream>>>(x, W1, nullptr, bufH, N, 0);
    k_selfloop<<<gNF4, b256, 0, stream>>>(bufH, dinv, bufA, N);
    k_scatter<<<gEw, b256, 0, stream>>>(bufH, srcp, dstp, dinv, bufA, E);
    k_bias_relu<<<gNF, b256, 0, stream>>>(bufA, b1, bufH, N);

    // ---- layer 2: h2 = relu(Agg(h1 @ W2) + b2) ----
    k_gemm_wmma<<<gGemm, b256, 0, stream>>>(bufH, W2, nullptr, bufA, N, 0);
    k_selfloop<<<gNF4, b256, 0, stream>>>(bufA, dinv, bufH, N);
    k_scatter<<<gEw, b256, 0, stream>>>(bufA, srcp, dstp, dinv, bufH, E);
    k_bias_relu<<<gNF, b256, 0, stream>>>(bufH, b2, bufA, N);

    // ---- decoder: out = relu(h2 @ Wd + bd) (bias+relu fused in epilogue) ----
    k_gemm_wmma<<<gGemm, b256, 0, stream>>>(bufA, Wd, bd, out, N, 1);
}